// GIN_test_50869592655530
// MI455X (gfx1250) — compile-verified
//
#include <hip/hip_runtime.h>
#include <hip/hip_bf16.h>

typedef float v2f __attribute__((ext_vector_type(2)));
typedef float v8f __attribute__((ext_vector_type(8)));

#define NNODES 50000
#define NEDGES 600000
#define NGRAPHS 512
#define HDIM 128
#define BN_EPS 1e-5f

// ---------------------------------------------------------------------------
// Elementwise helpers
// ---------------------------------------------------------------------------
__global__ void zero_kernel(float* __restrict__ p, int n) {
    int i = blockIdx.x * blockDim.x + threadIdx.x;
    if (i < n) p[i] = 0.0f;
}

// acc[n,h] = (1+eps) * emb[x_idx[n], h]
__global__ void init_agg_kernel(const int* __restrict__ x_idx,
                                const float* __restrict__ emb,
                                const float* __restrict__ eps,
                                float* __restrict__ acc) {
    int idx = blockIdx.x * blockDim.x + threadIdx.x;
    if (idx >= NNODES * HDIM) return;
    int node = idx >> 7, c = idx & 127;
    acc[idx] = (1.0f + eps[0]) * emb[x_idx[node] * HDIM + c];
}

// acc[dst[e], :] += emb[x_idx[src[e]], :]   (one thread = one edge x 4 columns)
__global__ void edge_scatter_kernel(const int* __restrict__ ei,
                                    const int* __restrict__ x_idx,
                                    const float* __restrict__ emb,
                                    float* __restrict__ acc) {
    int idx = blockIdx.x * blockDim.x + threadIdx.x;
    if (idx >= NEDGES * 32) return;
    int e = idx >> 5;
    int j = (idx & 31) << 2;
    int s = ei[e];
    int d = ei[NEDGES + e];
    const float4 v = *(const float4*)&emb[x_idx[s] * HDIM + j];
    float* p = &acc[d * HDIM + j];
    atomicAdd(p + 0, v.x);
    atomicAdd(p + 1, v.y);
    atomicAdd(p + 2, v.z);
    atomicAdd(p + 3, v.w);
}

// ---------------------------------------------------------------------------
// GEMM: Y[M,128] = A[M,128] @ W[128,128] + bias    (fp32 WMMA 16x16x4)
// 512 threads = 16 waves; wave (rs, cb): rows [strip*32 + rs*16, +16),
// cols [cb*16, +16). B operand lives in 64 VGPRs per lane, loaded once.
// ---------------------------------------------------------------------------
__global__ __launch_bounds__(512)
void gemm_kernel(const float* __restrict__ A, const float* __restrict__ W,
                 const float* __restrict__ bias, float* __restrict__ Y, int M) {
    __shared__ float Alds[32 * 132];   // 32 rows, stride 132 (bank-conflict pad)

    const int tid    = threadIdx.x;
    const int wave   = tid >> 5;
    const int lane   = tid & 31;
    const int rowSub = wave >> 3;          // 0..1
    const int colBlk = wave & 7;           // 0..7
    const int n      = lane & 15;          // column within 16-col block
    const int hi     = lane >> 4;          // half-wave select
    const int kb     = hi << 1;            // K base offset within K4 step: 0 or 2

    // Preload B fragments: W[k + kb + {0,1}][colBlk*16 + n] for k = 0,4,...,124
    v2f Breg[32];
#pragma unroll
    for (int t = 0; t < 32; ++t) {
        int k = t * 4 + kb;
        Breg[t].x = W[(k + 0) * HDIM + colBlk * 16 + n];
        Breg[t].y = W[(k + 1) * HDIM + colBlk * 16 + n];
    }
    const float bcol = bias[colBlk * 16 + n];

    const int nStrips = (M + 31) >> 5;
    for (int s = blockIdx.x; s < nStrips; s += gridDim.x) {
        const int m0 = s << 5;
        // cooperative load of 32x128 A tile
#pragma unroll
        for (int t = 0; t < 8; ++t) {
            int idx = tid + t * 512;       // 0..4095
            int r = idx >> 7, c = idx & 127;
            int gr = m0 + r;
            Alds[r * 132 + c] = (gr < M) ? A[gr * HDIM + c] : 0.0f;
        }
        __syncthreads();

        v8f acc = {};
        const float* ap = &Alds[(rowSub * 16 + (lane & 15)) * 132 + kb];
#pragma unroll
        for (int t = 0; t < 32; ++t) {
            v2f a;
            a.x = ap[t * 4 + 0];
            a.y = ap[t * 4 + 1];
            acc = __builtin_amdgcn_wmma_f32_16x16x4_f32(
                false, a, false, Breg[t], (short)0, acc, false, false);
        }

        // D layout: VGPR i -> row (i + hi*8), col = lane&15 within tile
#pragma unroll
        for (int i = 0; i < 8; ++i) {
            int orow = m0 + rowSub * 16 + i + (hi << 3);
            if (orow < M)
                Y[orow * HDIM + colBlk * 16 + n] = acc[i] + bcol;
        }
        __syncthreads();
    }
}

// ---------------------------------------------------------------------------
// Column stats: stats[c] = sum over rows, stats[128+c] = sum of squares
// ---------------------------------------------------------------------------
__global__ __launch_bounds__(256)
void colstats_kernel(const float* __restrict__ Y, int M, float* __restrict__ stats) {
    __shared__ float sdata[512];
    const int tid = threadIdx.x;
    const int c = tid & 127;
    const int half = tid >> 7;
    float s = 0.0f, ss = 0.0f;
    for (int r = blockIdx.x * 2 + half; r < M; r += gridDim.x * 2) {
        float v = Y[r * HDIM + c];
        s += v;
        ss += v * v;
    }
    sdata[tid] = s;
    sdata[256 + tid] = ss;
    __syncthreads();
    if (tid < 128) {
        atomicAdd(&stats[c], sdata[tid] + sdata[tid + 128]);
        atomicAdd(&stats[128 + c], sdata[256 + tid] + sdata[256 + tid + 128]);
    }
}

// y = relu(g * rsqrt(var+eps) * (y - mean) + be), training-mode batch stats
__global__ void bnrelu_kernel(float* __restrict__ Y, int M,
                              const float* __restrict__ stats,
                              const float* __restrict__ g,
                              const float* __restrict__ be) {
    int idx = blockIdx.x * blockDim.x + threadIdx.x;
    if (idx >= M * HDIM) return;
    int c = idx & 127;
    float invM = 1.0f / (float)M;
    float mean = stats[c] * invM;
    float var = stats[128 + c] * invM - mean * mean;
    float sc = g[c] * rsqrtf(var + BN_EPS);
    float v = sc * (Y[idx] - mean) + be[c];
    Y[idx] = v > 0.0f ? v : 0.0f;
}

// ---------------------------------------------------------------------------
// Global mean pool
// ---------------------------------------------------------------------------
__global__ void pool_scatter_kernel(const int* __restrict__ batch,
                                    const float* __restrict__ h,
                                    float* __restrict__ gsum) {
    int idx = blockIdx.x * blockDim.x + threadIdx.x;
    if (idx >= NNODES * HDIM) return;
    int node = idx >> 7, c = idx & 127;
    atomicAdd(&gsum[batch[node] * HDIM + c], h[idx]);
}

__global__ void pool_count_kernel(const int* __restrict__ batch, float* __restrict__ gcnt) {
    int n = blockIdx.x * blockDim.x + threadIdx.x;
    if (n < NNODES) atomicAdd(&gcnt[batch[n]], 1.0f);
}

__global__ void pool_div_kernel(const float* __restrict__ gsum,
                                const float* __restrict__ gcnt,
                                float* __restrict__ pooled) {
    int idx = blockIdx.x * blockDim.x + threadIdx.x;
    if (idx >= NGRAPHS * HDIM) return;
    int g = idx >> 7;
    pooled[idx] = gsum[idx] / fmaxf(gcnt[g], 1.0f);
}

// ---------------------------------------------------------------------------
// Final projection: out[g] = h[g,:] . fw + fb
// ---------------------------------------------------------------------------
__global__ __launch_bounds__(128)
void final_kernel(const float* __restrict__ h, const float* __restrict__ fw,
                  const float* __restrict__ fb, float* __restrict__ out) {
    __shared__ float red[128];
    int g = blockIdx.x, t = threadIdx.x;
    red[t] = h[g * HDIM + t] * fw[t];
    __syncthreads();
    for (int s = 64; s > 0; s >>= 1) {
        if (t < s) red[t] += red[t + s];
        __syncthreads();
    }
    if (t == 0) out[g] = red[0] + fb[0];
}

// ---------------------------------------------------------------------------
extern "C" void kernel_launch(void* const* d_in, const int* in_sizes, int n_in,
                              void* d_out, int out_size, void* d_ws, size_t ws_size,
                              hipStream_t stream) {
    const int*   x_idx = (const int*)d_in[0];
    const int*   ei    = (const int*)d_in[1];   // [2, NEDGES]
    const int*   batch = (const int*)d_in[2];
    const float* emb   = (const float*)d_in[3];
    const float* eps   = (const float*)d_in[4];
    const float* P[24];
    for (int i = 0; i < 24; ++i) P[i] = (const float*)d_in[5 + i];
    // P index map: [c_w1,c_b1,c_g1,c_be1, c_w2,c_b2,c_g2,c_be2,
    //               l1_w1..., l1_w2..., l2_w1..., l2_w2...]
    const float* fw = (const float*)d_in[29];
    const float* fb = (const float*)d_in[30];
    float* out = (float*)d_out;

    float* ws   = (float*)d_ws;
    float* bufA = ws;                       // NNODES*HDIM
    float* bufB = bufA + (size_t)NNODES * HDIM;
    float* gsum = bufB + (size_t)NNODES * HDIM;  // NGRAPHS*HDIM
    float* gcnt = gsum + NGRAPHS * HDIM;         // NGRAPHS
    float* p0   = gcnt + NGRAPHS;                // NGRAPHS*HDIM
    float* p1   = p0 + NGRAPHS * HDIM;           // NGRAPHS*HDIM
    float* stats = p1 + NGRAPHS * HDIM;          // 256

    const int bigElems = NNODES * HDIM;

    auto gemm = [&](const float* A, const float* W, const float* b, float* Y, int M) {
        int strips = (M + 31) >> 5;
        int grid = strips < 640 ? strips : 640;
        gemm_kernel<<<grid, 512, 0, stream>>>(A, W, b, Y, M);
    };
    auto bn = [&](float* Y, int M, const float* g, const float* be) {
        zero_kernel<<<1, 256, 0, stream>>>(stats, 256);
        int sgrid = (M + 1) >> 1;
        if (sgrid > 512) sgrid = 512;
        colstats_kernel<<<sgrid, 256, 0, stream>>>(Y, M, stats);
        bnrelu_kernel<<<(M * HDIM + 255) / 256, 256, 0, stream>>>(Y, M, stats, g, be);
    };

    // 1) x = emb[x_idx]; agg = (1+eps)*x + segment_sum(x[src], dst)
    init_agg_kernel<<<(bigElems + 255) / 256, 256, 0, stream>>>(x_idx, emb, eps, bufA);
    edge_scatter_kernel<<<(NEDGES * 32 + 255) / 256, 256, 0, stream>>>(ei, x_idx, emb, bufA);

    // 2) GINConv MLP on nodes
    gemm(bufA, P[0], P[1], bufB, NNODES);  bn(bufB, NNODES, P[2], P[3]);
    gemm(bufB, P[4], P[5], bufA, NNODES);  bn(bufA, NNODES, P[6], P[7]);

    // 3) global mean pool
    zero_kernel<<<(NGRAPHS * HDIM + 255) / 256, 256, 0, stream>>>(gsum, NGRAPHS * HDIM);
    zero_kernel<<<(NGRAPHS + 255) / 256, 256, 0, stream>>>(gcnt, NGRAPHS);
    pool_scatter_kernel<<<(bigElems + 255) / 256, 256, 0, stream>>>(batch, bufA, gsum);
    pool_count_kernel<<<(NNODES + 255) / 256, 256, 0, stream>>>(batch, gcnt);
    pool_div_kernel<<<(NGRAPHS * HDIM + 255) / 256, 256, 0, stream>>>(gsum, gcnt, p0);

    // 4) head MLPs (relu(_mlp(..)) == _mlp(..) since _mlp ends in relu)
    gemm(p0, P[8],  P[9],  p1, NGRAPHS);  bn(p1, NGRAPHS, P[10], P[11]);
    gemm(p1, P[12], P[13], p0, NGRAPHS);  bn(p0, NGRAPHS, P[14], P[15]);
    gemm(p0, P[16], P[17], p1, NGRAPHS);  bn(p1, NGRAPHS, P[18], P[19]);
    gemm(p1, P[20], P[21], p0, NGRAPHS);  bn(p0, NGRAPHS, P[22], P[23]);

    // 5) final linear
    final_kernel<<<NGRAPHS, 128, 0, stream>>>(p0, fw, fb, out);
}